// FractalMicroTransformRungs_76888504533224
// MI455X (gfx1250) — compile-verified
//
#include <hip/hip_runtime.h>
#include <math.h>

typedef __attribute__((ext_vector_type(2))) float v2f;
typedef __attribute__((ext_vector_type(4))) float v4f;
typedef __attribute__((ext_vector_type(8))) float v8f;

#define SO3_EPS 1e-5f
#define CLAMPV 10000.0f

// ---------------------------------------------------------------------------
// Prologue: fold SO(4) rotation into the first linear layer.
// A1 = W1 @ (R(q) @ L(p))   (4x4, row-major) -> workspace
// ---------------------------------------------------------------------------
__global__ void fmr_prep(const float* __restrict__ so3_p,
                         const float* __restrict__ so3_q,
                         const float* __restrict__ W1,
                         float* __restrict__ A1out) {
  if (blockIdx.x != 0 || threadIdx.x != 0) return;

  float P[4], Q[4];
  {
    float v0 = so3_p[0], v1 = so3_p[1], v2 = so3_p[2];
    float ang = sqrtf(v0 * v0 + v1 * v1 + v2 * v2);
    float safe = fmaxf(ang, SO3_EPS);
    float s = sinf(0.5f * ang) / safe;
    if (ang < SO3_EPS) { P[0] = 1.f; P[1] = P[2] = P[3] = 0.f; }
    else { P[0] = cosf(0.5f * ang); P[1] = v0 * s; P[2] = v1 * s; P[3] = v2 * s; }
  }
  {
    float v0 = so3_q[0], v1 = so3_q[1], v2 = so3_q[2];
    float ang = sqrtf(v0 * v0 + v1 * v1 + v2 * v2);
    float safe = fmaxf(ang, SO3_EPS);
    float s = sinf(0.5f * ang) / safe;
    if (ang < SO3_EPS) { Q[0] = 1.f; Q[1] = Q[2] = Q[3] = 0.f; }
    else { Q[0] = cosf(0.5f * ang); Q[1] = v0 * s; Q[2] = v1 * s; Q[3] = v2 * s; }
  }

  // left-multiplication matrix: p*x = L(p) x   (w,x,y,z column vectors)
  float L[16] = { P[0], -P[1], -P[2], -P[3],
                  P[1],  P[0], -P[3],  P[2],
                  P[2],  P[3],  P[0], -P[1],
                  P[3], -P[2],  P[1],  P[0] };
  // right-multiplication matrix: x*q = R(q) x
  float R[16] = { Q[0], -Q[1], -Q[2], -Q[3],
                  Q[1],  Q[0],  Q[3], -Q[2],
                  Q[2], -Q[3],  Q[0],  Q[1],
                  Q[3],  Q[2], -Q[1],  Q[0] };

  float M[16];
  for (int i = 0; i < 4; ++i)
    for (int j = 0; j < 4; ++j) {
      float acc = 0.f;
      for (int k = 0; k < 4; ++k) acc += R[i * 4 + k] * L[k * 4 + j];
      M[i * 4 + j] = acc;
    }
  for (int i = 0; i < 4; ++i)
    for (int j = 0; j < 4; ++j) {
      float acc = 0.f;
      for (int k = 0; k < 4; ++k) acc += W1[i * 4 + k] * M[k * 4 + j];
      A1out[i * 4 + j] = acc;
    }
}

// ---------------------------------------------------------------------------
// Branchless exact GELU: 0.5*x*(1+erf(x/sqrt2)) with Abramowitz-Stegun 7.1.26
// erf (|err|<=1.5e-7). One v_rcp_f32 + one native v_exp_f32, no branches.
// ---------------------------------------------------------------------------
__device__ __forceinline__ float gelu_exact(float x) {
  const float z = __builtin_fabsf(x) * 0.70710678118654752f;
  const float t = __builtin_amdgcn_rcpf(__builtin_fmaf(0.3275911f, z, 1.0f));
  float p = __builtin_fmaf(1.061405429f, t, -1.453152027f);
  p = __builtin_fmaf(p, t, 1.421413741f);
  p = __builtin_fmaf(p, t, -0.284496736f);
  p = __builtin_fmaf(p, t, 0.254829592f);
  p *= t;
  const float e  = __expf(-z * z);                       // v_exp_f32
  const float er = __builtin_copysignf(__builtin_fmaf(-p, e, 1.0f), x);
  return 0.5f * x * (1.0f + er);
}

__device__ __forceinline__ float swapx16(float x) {
  // ds_swizzle group-of-32: xor=0x10, or=0, and=0x1f  (SWAPX16)
  return __int_as_float(__builtin_amdgcn_ds_swizzle(__float_as_int(x), 0x401F));
}

__device__ __forceinline__ float clampv(float x) {
  return fminf(fmaxf(x, -CLAMPV), CLAMPV);
}

// ---------------------------------------------------------------------------
// Main kernel: 32 rows per wave-iteration via 4x V_WMMA_F32_16X16X4_F32.
// Transposed formulation: D[m][n] = component m of batch row n.
// Group A = rows g*32+0..15, group B = rows g*32+16..31.
// ---------------------------------------------------------------------------
__global__ void __launch_bounds__(256)
fmr_main(const float* __restrict__ x,     // [B,4]
         const float* __restrict__ sc,    // [B,4]
         const float* __restrict__ b1v,   // [4]
         const float* __restrict__ b2v,   // [4]
         const float* __restrict__ W2,    // [4,4]
         const float* __restrict__ A1,    // [4,4] from workspace
         float* __restrict__ out,         // [B,4]
         int ngroups32, int nrows) {
  const int lane = threadIdx.x & 31;
  const int half = lane >> 4;   // 0: lanes 0-15, 1: lanes 16-31
  const int lm   = lane & 15;   // row-in-group / matrix-row index

  // A-operand layout (16x4 f32): lane L, VGPR j holds A[L%16][2*(L/16)+j].
  // Rows >= 4 are zero-padded so the unused region of D stays exactly 0.
  v2f aA = {0.f, 0.f}, aB = {0.f, 0.f};
  if (lm < 4) {
    aA[0] = A1[lm * 4 + 2 * half + 0];
    aA[1] = A1[lm * 4 + 2 * half + 1];
    aB[0] = W2[lm * 4 + 2 * half + 0];
    aB[1] = W2[lm * 4 + 2 * half + 1];
  }

  // Stage-1 accumulator = b1 broadcast in C layout (VGPR m, lanes 0-15 = M=m).
  v8f c1 = {0.f, 0.f, 0.f, 0.f, 0.f, 0.f, 0.f, 0.f};
  if (half == 0) { c1[0] = b1v[0]; c1[1] = b1v[1]; c1[2] = b1v[2]; c1[3] = b1v[3]; }
  float bb[4] = {b2v[0], b2v[1], b2v[2], b2v[3]};

  const int wave   = (int)((blockIdx.x * blockDim.x + threadIdx.x) >> 5);
  const int nwaves = (int)((gridDim.x * blockDim.x) >> 5);

  for (int g = wave; g < ngroups32; g += nwaves) {
    const int base  = g * 32;
    const int rowA  = base + lm;        // group A row for this lane's lm
    const int rowB  = base + 16 + lm;   // group B row
    const int rowS  = base + lane;      // scaffold/store row (full wave)
    const int ldA   = (rowA < nrows) ? rowA : (nrows - 1);
    const int ldB   = (rowB < nrows) ? rowB : (nrows - 1);
    const int ldS   = (rowS < nrows) ? rowS : (nrows - 1);

    // ---- all global loads up front (one wait before the WMMAs) ----
    // B-operand = x^T (4x16): lane L b64-loads comps {2*half, 2*half+1}.
    v2f bxA = *(const v2f*)(x + (size_t)ldA * 4 + 2 * half);
    v2f bxB = *(const v2f*)(x + (size_t)ldB * 4 + 2 * half);
    // scaffold: full-wave coalesced b128, lane L holds row base+L.
    v4f s = *(const v4f*)(sc + (size_t)ldS * 4);

    // ---- stage 1: pre^T = A1 * x^T + b1 (both groups) ----
    v8f d1A = __builtin_amdgcn_wmma_f32_16x16x4_f32(
        false, aA, false, bxA, (short)0, c1, false, false);
    v8f d1B = __builtin_amdgcn_wmma_f32_16x16x4_f32(
        false, aA, false, bxB, (short)0, c1, false, false);

    // exact GELU on the 4 valid component planes (invalid slots are 0 -> 0)
    d1A[0] = gelu_exact(d1A[0]);  d1B[0] = gelu_exact(d1B[0]);
    d1A[1] = gelu_exact(d1A[1]);  d1B[1] = gelu_exact(d1B[1]);
    d1A[2] = gelu_exact(d1A[2]);  d1B[2] = gelu_exact(d1B[2]);
    d1A[3] = gelu_exact(d1A[3]);  d1B[3] = gelu_exact(d1B[3]);

    // Re-pack h^T into B-operand layout: K0|K2 in VGPR0, K1|K3 in VGPR1.
    v2f bhA, bhB;
    {
      const float a2 = swapx16(d1A[2]);
      const float a3 = swapx16(d1A[3]);
      const float b2_ = swapx16(d1B[2]);
      const float b3_ = swapx16(d1B[3]);
      bhA[0] = half ? a2 : d1A[0];
      bhA[1] = half ? a3 : d1A[1];
      bhB[0] = half ? b2_ : d1B[0];
      bhB[1] = half ? b3_ : d1B[1];
    }

    // Stage-2 accumulators = scaffold^T + b2 in C layout.
    // Group A rows live in lanes 0-15 of s; group B rows in lanes 16-31 ->
    // swizzle them down to lanes 0-15.
    v8f cA = {0.f, 0.f, 0.f, 0.f, 0.f, 0.f, 0.f, 0.f};
    v8f cB = {0.f, 0.f, 0.f, 0.f, 0.f, 0.f, 0.f, 0.f};
#pragma unroll
    for (int r = 0; r < 4; ++r) {
      const float sB = swapx16(s[r]);
      cA[r] = (half == 0) ? (s[r] + bb[r]) : 0.f;
      cB[r] = (half == 0) ? (sB + bb[r]) : 0.f;
    }

    // ---- stage 2: out^T = W2 * h^T + (scaffold^T + b2) ----
    v8f d2A = __builtin_amdgcn_wmma_f32_16x16x4_f32(
        false, aB, false, bhA, (short)0, cA, false, false);
    v8f d2B = __builtin_amdgcn_wmma_f32_16x16x4_f32(
        false, aB, false, bhB, (short)0, cB, false, false);

    // ---- merge + clamp + one full-wave coalesced b128 store ----
    // Group A row n in lanes n (<16); group B row n swizzled up to lane n+16.
    v4f o;
#pragma unroll
    for (int r = 0; r < 4; ++r) {
      const float up = swapx16(d2B[r]);       // lanes 16-31 <- group B
      o[r] = clampv(half ? up : d2A[r]);
    }
    if (rowS < nrows) {
      *(v4f*)(out + (size_t)rowS * 4) = o;
    }
  }
}

// ---------------------------------------------------------------------------
extern "C" void kernel_launch(void* const* d_in, const int* in_sizes, int n_in,
                              void* d_out, int out_size, void* d_ws, size_t ws_size,
                              hipStream_t stream) {
  const float* x     = (const float*)d_in[0];  // main_tan [B,4]
  const float* sc    = (const float*)d_in[1];  // scaffold_modulation [B,4]
  const float* so3_p = (const float*)d_in[2];  // [3]
  const float* so3_q = (const float*)d_in[3];  // [3]
  const float* W1    = (const float*)d_in[4];  // [4,4]
  const float* b1    = (const float*)d_in[5];  // [4]
  const float* W2    = (const float*)d_in[6];  // [4,4]
  const float* b2    = (const float*)d_in[7];  // [4]
  float* out = (float*)d_out;
  float* A1  = (float*)d_ws;                   // 16 floats

  const int nrows     = in_sizes[0] / 4;
  const int ngroups32 = (nrows + 31) / 32;

  fmr_prep<<<1, 32, 0, stream>>>(so3_p, so3_q, W1, A1);

  const int threads = 256;                 // 8 waves/block
  int blocks = (ngroups32 + 7) / 8;        // one 32-row group per wave/sweep
  if (blocks > 4096) blocks = 4096;        // grid-stride beyond this
  if (blocks < 1) blocks = 1;
  fmr_main<<<blocks, threads, 0, stream>>>(x, sc, b1, b2, W2, A1, out,
                                           ngroups32, nrows);
}